// mask_ssm_62818191671675
// MI455X (gfx1250) — compile-verified
//
#include <hip/hip_runtime.h>
#include <math.h>

typedef __attribute__((ext_vector_type(2))) float v2f;
typedef __attribute__((ext_vector_type(8))) float v8f;

#define B_    8
#define CIN   96
#define H_    192
#define W_    192
#define HW    36864
#define CH2   384
#define DIN   192
#define LSEL  2560
#define K4    4
#define NST   16
#define DTR   6
#define CPROJ 38        // DTR + 2*NST
#define GC    12        // cells per side
#define NCELL 144
#define LN_EPS 1e-5f
#define SCH   16        // scan LDS chunk

// ---- workspace layout (float offsets) ----
#define N_XH   ((size_t)B_*DIN*HW)          // 56,623,104
#define F_XH   ((size_t)0)
#define F_Z    (F_XH + N_XH)
#define F_YBUF (F_Z + N_XH)
#define N_Y    ((size_t)B_*K4*LSEL*DIN)     // 15,728,640
#define F_Y    (F_YBUF + N_XH)
#define F_DTS  (F_Y + N_Y)
#define N_XS   ((size_t)B_*DIN*LSEL)        // 3,932,160
#define F_XS   (F_DTS + N_Y)
#define N_XDBL ((size_t)B_*K4*CPROJ*LSEL)   // 3,112,960
#define F_XDBL (F_XS + N_XS)
#define F_YSC  (F_XDBL + N_XDBL)
#define F_CELLS (F_YSC + N_XS)
#define F_DALL (F_CELLS + (size_t)B_*NCELL)
#define F_SEL  (F_DALL + (size_t)(K4+1)*DIN)   // int region, 312 ints per batch

__device__ __forceinline__ float silu_f(float v) { return v / (1.f + __expf(-v)); }

// selinfo per batch (312 ints): [0..143] cmask, [144..155] rowSelCells,
// [156..167] rowSelPrefix, [168..311] colPrefix
__device__ __forceinline__ void pix_rank(const int* __restrict__ si, int h, int w,
                                         int& sel, int& rank) {
  int cr = h >> 4, cc = w >> 4;
  int m = si[cr*GC + cc];
  int before = 16*(16*si[156 + cr] + (h & 15)*si[144 + cr]);
  before += 16*si[168 + cr*GC + cc] + (m ? (w & 15) : 0);
  sel = m; rank = before;
}

// ---------------- K0: Ds_all = silu(proj_d_w @ Ds) ----------------
__global__ void k_dall(const float* __restrict__ pw, const float* __restrict__ Ds,
                       float* __restrict__ Dall) {
  int i = blockIdx.x * blockDim.x + threadIdx.x;
  if (i >= (K4+1)*DIN) return;
  const float* row = pw + (size_t)i*DIN;
  float s = 0.f;
  for (int j = 0; j < DIN; j++) s += row[j]*Ds[j];
  Dall[i] = silu_f(s);
}

// ------- implicit-GEMM 3x3 conv via WMMA f32 16x16x4, K reordered as (tap, c) ----
// Each wave: 32 output rows (2 x 16) x 64 pixels (4 x 16). Padding by clamp+mask.
// Inner loop: issue all 12 loads into distinct registers first, then 8 WMMAs,
// with pointer-bump addressing only (no per-load 64-bit address chains).
template<int CINX, int COUT, bool ACT>
__global__ void k_conv3x3(const float* __restrict__ x, const float* __restrict__ wt,
                          const float* __restrict__ bias,
                          float* __restrict__ out0, float* __restrict__ out1) {
  const int KTOT = CINX*9;
  const int MT = COUT/32;          // 32-row supertiles
  int wave = (int)((blockIdx.x * (size_t)blockDim.x + threadIdx.x) >> 5);
  int lane = threadIdx.x & 31;
  int mtile2 = wave % MT;
  int ntile = wave / MT;           // 64 pixels per wave
  int n0 = ntile * 64;
  int mlo = lane & 15;
  int koff = (lane >> 4) * 2;      // A/B lane-half K offset per ISA layout
  int m0 = mtile2*32 + mlo;        // rows for A fragment 0
  int m1 = m0 + 16;                // rows for A fragment 1

  int pb[4], ph[4], pw[4];
  #pragma unroll
  for (int j = 0; j < 4; j++) {
    int n = n0 + j*16 + mlo;
    pb[j] = n / HW; int rem = n - pb[j]*HW;
    ph[j] = rem / W_; pw[j] = rem - ph[j]*W_;
  }

  v8f acc[2][4] = {{v8f{}, v8f{}, v8f{}, v8f{}}, {v8f{}, v8f{}, v8f{}, v8f{}}};

  #pragma unroll
  for (int tap = 0; tap < 9; tap++) {
    const int r = tap / 3, s = tap % 3;           // compile-time constants
    // per-pixel clamped base pointer + validity mask for this tap (koff folded in)
    float m0f, m1f, m2f, m3f;
    const float *p0, *p1, *p2, *p3;
    {
      const float* bp[4]; float msk[4];
      #pragma unroll
      for (int j = 0; j < 4; j++) {
        int ih = ph[j] + r - 1, iw = pw[j] + s - 1;
        bool valid = ((unsigned)ih < (unsigned)H_) && ((unsigned)iw < (unsigned)W_);
        int ihc = min(max(ih, 0), H_-1), iwc = min(max(iw, 0), W_-1);
        msk[j] = valid ? 1.f : 0.f;
        bp[j] = x + ((size_t)pb[j]*CINX*H_ + ihc)*W_ + iwc + (size_t)koff*HW;
      }
      m0f = msk[0]; m1f = msk[1]; m2f = msk[2]; m3f = msk[3];
      p0 = bp[0]; p1 = bp[1]; p2 = bp[2]; p3 = bp[3];
    }
    const float* wp0 = wt + (size_t)m0*KTOT + tap + (size_t)koff*9;
    const float* wp1 = wt + (size_t)m1*KTOT + tap + (size_t)koff*9;

    for (int cc = 0; cc < CINX; cc += 4) {
      // ---- issue all loads first (independent destination registers) ----
      float a0x = wp0[0], a0y = wp0[9];
      float a1x = wp1[0], a1y = wp1[9];
      float bx0 = p0[0], by0 = p0[HW];
      float bx1 = p1[0], by1 = p1[HW];
      float bx2 = p2[0], by2 = p2[HW];
      float bx3 = p3[0], by3 = p3[HW];
      wp0 += 36; wp1 += 36;
      p0 += 4*HW; p1 += 4*HW; p2 += 4*HW; p3 += 4*HW;
      // ---- form fragments, run 8 WMMAs ----
      v2f a0 = {a0x, a0y};
      v2f a1 = {a1x, a1y};
      v2f b0 = {bx0*m0f, by0*m0f};
      v2f b1 = {bx1*m1f, by1*m1f};
      v2f b2 = {bx2*m2f, by2*m2f};
      v2f b3 = {bx3*m3f, by3*m3f};
      acc[0][0] = __builtin_amdgcn_wmma_f32_16x16x4_f32(false, a0, false, b0, (short)0, acc[0][0], false, false);
      acc[1][0] = __builtin_amdgcn_wmma_f32_16x16x4_f32(false, a1, false, b0, (short)0, acc[1][0], false, false);
      acc[0][1] = __builtin_amdgcn_wmma_f32_16x16x4_f32(false, a0, false, b1, (short)0, acc[0][1], false, false);
      acc[1][1] = __builtin_amdgcn_wmma_f32_16x16x4_f32(false, a1, false, b1, (short)0, acc[1][1], false, false);
      acc[0][2] = __builtin_amdgcn_wmma_f32_16x16x4_f32(false, a0, false, b2, (short)0, acc[0][2], false, false);
      acc[1][2] = __builtin_amdgcn_wmma_f32_16x16x4_f32(false, a1, false, b2, (short)0, acc[1][2], false, false);
      acc[0][3] = __builtin_amdgcn_wmma_f32_16x16x4_f32(false, a0, false, b3, (short)0, acc[0][3], false, false);
      acc[1][3] = __builtin_amdgcn_wmma_f32_16x16x4_f32(false, a1, false, b3, (short)0, acc[1][3], false, false);
    }
  }

  #pragma unroll
  for (int half = 0; half < 2; half++) {
    #pragma unroll
    for (int j = 0; j < 4; j++) {
      #pragma unroll
      for (int i = 0; i < 8; i++) {
        int o = mtile2*32 + half*16 + i + 8*(lane >> 4); // C/D: VGPR i -> row i/i+8
        float v = acc[half][j][i];
        if (ACT) {
          v += bias[o];
          v = silu_f(v);
          if (o < DIN)
            out0[(((size_t)pb[j]*DIN + o)*H_ + ph[j])*W_ + pw[j]] = v;
          else
            out1[(((size_t)pb[j]*DIN + (o - DIN))*H_ + ph[j])*W_ + pw[j]] = v;
        } else {
          out0[(((size_t)pb[j]*COUT + o)*H_ + ph[j])*W_ + pw[j]] = v;
        }
      }
    }
  }
}

// ---------------- K2: 16x16 cell means of sigmoid(channel-mean) ----------------
__global__ void k_cells(const float* __restrict__ xh, float* __restrict__ cells) {
  __shared__ float red[256];
  int cell = blockIdx.x;
  int b = cell / NCELL, ci = cell % NCELL;
  int cr = ci / GC, cc = ci % GC;
  int t = threadIdx.x;
  int ph = cr*16 + (t >> 4), pw = cc*16 + (t & 15);
  const float* base = xh + (size_t)b*DIN*HW + (size_t)ph*W_ + pw;
  float s = 0.f;
  for (int d = 0; d < DIN; d++) s += base[(size_t)d*HW];
  s = 1.f/(1.f + __expf(-(s/(float)DIN)));
  red[t] = s; __syncthreads();
  for (int o = 128; o > 0; o >>= 1) {
    if (t < o) red[t] += red[t+o];
    __syncthreads();
  }
  if (t == 0) cells[cell] = red[0]/256.f;
}

// ---------------- K3: top-10 threshold + prefix tables ----------------
__global__ void k_select(const float* __restrict__ cells, int* __restrict__ selinfo) {
  __shared__ float sc[NCELL];
  __shared__ unsigned minv_bits;
  int b = blockIdx.x, t = threadIdx.x;
  if (t < NCELL) sc[t] = cells[b*NCELL + t];
  if (t == 0) minv_bits = 0xFFFFFFFFu;
  __syncthreads();
  if (t < NCELL) {
    float v = sc[t];
    int rank = 0;
    for (int j = 0; j < NCELL; j++) {
      float u = sc[j];
      if (u > v || (u == v && j < t)) rank++;
    }
    if (rank < 10) atomicMin(&minv_bits, __float_as_uint(v)); // sigmoid outputs > 0
  }
  __syncthreads();
  float minv = __uint_as_float(minv_bits);
  int* si = selinfo + b*312;
  if (t < NCELL) si[t] = (sc[t] >= minv) ? 1 : 0;
  __syncthreads();
  if (t < GC) {
    int s = 0;
    for (int cc = 0; cc < GC; cc++) { si[168 + t*GC + cc] = s; s += si[t*GC + cc]; }
    si[144 + t] = s;
  }
  __syncthreads();
  if (t == 0) {
    int s = 0;
    for (int cr = 0; cr < GC; cr++) { si[156 + cr] = s; s += si[144 + cr]; }
  }
}

// ---------------- K4: gather selected pixels -> xs_row (b,d,l) ----------------
__global__ void k_gather(const float* __restrict__ xh, const int* __restrict__ selinfo,
                         float* __restrict__ xs) {
  int p = blockIdx.x * blockDim.x + threadIdx.x;
  if (p >= B_*HW) return;
  int b = p / HW, pp = p % HW;
  int h = pp / W_, w = pp % W_;
  int sel, l;
  pix_rank(selinfo + b*312, h, w, sel, l);
  if (!sel || l >= LSEL) return;
  const float* src = xh + (size_t)b*DIN*HW + pp;
  float* dst = xs + (size_t)b*DIN*LSEL + l;
  for (int d = 0; d < DIN; d++) dst[(size_t)d*LSEL] = src[(size_t)d*HW];
}

// ---------------- K5a: x_dbl[b,k,c,l] = sum_d xpw[k,c,d]*xs[b,k,d,l] ----------------
__global__ void k_xdbl(const float* __restrict__ xs, const float* __restrict__ xpw,
                       float* __restrict__ xdbl) {
  size_t i = blockIdx.x * (size_t)blockDim.x + threadIdx.x;
  if (i >= (size_t)B_*K4*CPROJ*LSEL) return;
  int l = (int)(i % LSEL); size_t r = i / LSEL;
  int c = (int)(r % CPROJ); r /= CPROJ;
  int k = (int)(r % K4);  int b = (int)(r / K4);
  int ls = (k < 2) ? l : (LSEL - 1 - l);     // directions 2,3 = reversed sequence
  const float* xb = xs + (size_t)b*DIN*LSEL + ls;
  const float* wv = xpw + ((size_t)k*CPROJ + c)*DIN;
  float s = 0.f;
  for (int d = 0; d < DIN; d++) s += wv[d]*xb[(size_t)d*LSEL];
  xdbl[i] = s;
}

// ---------------- K5b: dts[b,k,d,l] = sum_r x_dbl[b,k,r,l]*dtw[k,d,r] ----------------
__global__ void k_dts(const float* __restrict__ xdbl, const float* __restrict__ dtw,
                      float* __restrict__ dts) {
  size_t i = blockIdx.x * (size_t)blockDim.x + threadIdx.x;
  if (i >= (size_t)B_*K4*DIN*LSEL) return;
  int l = (int)(i % LSEL); size_t r = i / LSEL;
  int d = (int)(r % DIN); r /= DIN;
  int k = (int)(r % K4);  int b = (int)(r / K4);
  const float* xc = xdbl + (((size_t)b*K4 + k)*CPROJ)*LSEL + l;
  const float* wv = dtw + ((size_t)k*DIN + d)*DTR;
  float s = 0.f;
  #pragma unroll
  for (int r2 = 0; r2 < DTR; r2++) s += wv[r2]*xc[(size_t)r2*LSEL];
  dts[i] = s;
}

// ---------------- K6: selective scan, one block per (b,k), thread = channel d ----
__global__ void k_scan(const float* __restrict__ xs, const float* __restrict__ xdbl,
                       const float* __restrict__ dts, const float* __restrict__ dtb,
                       const float* __restrict__ A_logs, float* __restrict__ Y) {
  __shared__ float su [DIN*(SCH+1)];
  __shared__ float sdt[DIN*(SCH+1)];
  __shared__ float sy [DIN*(SCH+1)];
  __shared__ float sB [NST*(SCH+1)];
  __shared__ float sC [NST*(SCH+1)];
  int bk = blockIdx.x;
  int b = bk / K4, k = bk % K4;
  int d = threadIdx.x;  // 0..191
  float A[NST], st[NST];
  #pragma unroll
  for (int n = 0; n < NST; n++) {
    A[n] = -__expf(A_logs[((size_t)k*DIN + d)*NST + n]);
    st[n] = 0.f;
  }
  float bias = dtb[k*DIN + d];
  const float* dtrow = dts + (((size_t)b*K4 + k)*DIN + d)*LSEL;
  const float* xrow  = xs  + ((size_t)b*DIN + d)*LSEL;
  const float* Bbase = xdbl + (((size_t)b*K4 + k)*CPROJ + DTR)*LSEL;
  const float* Cbase = Bbase + (size_t)NST*LSEL;
  float* Yb = Y + ((size_t)b*K4 + k)*LSEL*DIN;
  bool rev = (k >= 2);

  for (int l0 = 0; l0 < LSEL; l0 += SCH) {
    for (int j = 0; j < SCH; j++) {
      int l = l0 + j;
      sdt[d*(SCH+1)+j] = dtrow[l];
      su [d*(SCH+1)+j] = xrow[rev ? (LSEL-1-l) : l];
    }
    if (d < NST) {
      for (int j = 0; j < SCH; j++) {
        sB[d*(SCH+1)+j] = Bbase[(size_t)d*LSEL + l0 + j];
        sC[d*(SCH+1)+j] = Cbase[(size_t)d*LSEL + l0 + j];
      }
    }
    __syncthreads();
    for (int j = 0; j < SCH; j++) {
      float dtv = sdt[d*(SCH+1)+j] + bias;
      dtv = (dtv > 20.f) ? dtv : log1pf(__expf(dtv));  // softplus
      float du = dtv * su[d*(SCH+1)+j];
      float y = 0.f;
      #pragma unroll
      for (int n = 0; n < NST; n++) {
        st[n] = __expf(dtv*A[n])*st[n] + du*sB[n*(SCH+1)+j];
        y += st[n]*sC[n*(SCH+1)+j];
      }
      sy[d*(SCH+1)+j] = y;
    }
    __syncthreads();
    for (int j = 0; j < SCH; j++)
      Yb[(size_t)(l0+j)*DIN + d] = sy[d*(SCH+1)+j];   // coalesced over d
    __syncthreads();
  }
}

// ---------------- K7: combine 4 directions -> ysum (b,l,d) ----------------
__global__ void k_comb(const float* __restrict__ Y, float* __restrict__ ysc) {
  size_t i = blockIdx.x * (size_t)blockDim.x + threadIdx.x;
  if (i >= (size_t)B_*LSEL*DIN) return;
  int d = (int)(i % DIN); size_t r = i / DIN;
  int l = (int)(r % LSEL); int b = (int)(r / LSEL);
  int lr = LSEL - 1 - l;
  size_t base = (size_t)b*K4*LSEL*DIN;
  ysc[i] = Y[base + ((size_t)0*LSEL + l )*DIN + d]
         + Y[base + ((size_t)1*LSEL + l )*DIN + d]
         + Y[base + ((size_t)2*LSEL + lr)*DIN + d]
         + Y[base + ((size_t)3*LSEL + lr)*DIN + d];
}

// ---------------- K8: assemble + LayerNorm + gate (block = pixel, thread = d) ---
__global__ void k_ln(const float* __restrict__ xh, const float* __restrict__ z,
                     const float* __restrict__ ysc, const float* __restrict__ Dall,
                     const int* __restrict__ selinfo,
                     const float* __restrict__ lnw, const float* __restrict__ lnb,
                     float* __restrict__ ybuf) {
  __shared__ float red[DIN];
  __shared__ float red2[DIN];
  int pid = blockIdx.x;
  int b = pid / HW, p = pid % HW;
  int h = p / W_, w = p % W_;
  int d = threadIdx.x;
  int sel, l;
  pix_rank(selinfo + b*312, h, w, sel, l);
  float xv = xh[((size_t)b*DIN + d)*HW + p];
  float v;
  if (sel && l < LSEL) {
    float Dsum = Dall[d] + Dall[DIN+d] + Dall[2*DIN+d] + Dall[3*DIN+d];
    v = 0.25f*(ysc[((size_t)b*LSEL + l)*DIN + d] + Dsum*xv);
  } else {
    v = Dall[4*DIN + d] * xv;      // Ds_b path for non-selected pixels
  }
  red[d] = v; red2[d] = v*v;
  __syncthreads();
  for (int o = 96; o >= 3; o >>= 1) {       // 96,48,24,12,6,3
    if (d < o) { red[d] += red[d+o]; red2[d] += red2[d+o]; }
    __syncthreads();
  }
  float mu  = (red[0] + red[1] + red[2]) / (float)DIN;
  float ms  = (red2[0] + red2[1] + red2[2]) / (float)DIN;
  float var = ms - mu*mu;
  float yn = (v - mu) * rsqrtf(var + LN_EPS) * lnw[d] + lnb[d];
  ybuf[((size_t)b*DIN + d)*HW + p] = yn * z[((size_t)b*DIN + d)*HW + p];
}

// ---------------- launch ----------------
extern "C" void kernel_launch(void* const* d_in, const int* in_sizes, int n_in,
                              void* d_out, int out_size, void* d_ws, size_t ws_size,
                              hipStream_t stream) {
  const float* x    = (const float*)d_in[0];
  const float* w1   = (const float*)d_in[1];
  const float* b1   = (const float*)d_in[2];
  const float* xpw  = (const float*)d_in[3];
  const float* dtw  = (const float*)d_in[4];
  const float* dtb  = (const float*)d_in[5];
  const float* alog = (const float*)d_in[6];
  const float* Ds   = (const float*)d_in[7];
  const float* pdw  = (const float*)d_in[8];
  const float* lnw  = (const float*)d_in[9];
  const float* lnb  = (const float*)d_in[10];
  const float* w2   = (const float*)d_in[11];
  float* out = (float*)d_out;

  float* ws = (float*)d_ws;
  float* xh   = ws + F_XH;
  float* zbuf = ws + F_Z;
  float* ybuf = ws + F_YBUF;
  float* Y    = ws + F_Y;
  float* dts  = ws + F_DTS;
  float* xs   = ws + F_XS;
  float* xdbl = ws + F_XDBL;
  float* ysc  = ws + F_YSC;
  float* cells = ws + F_CELLS;
  float* Dall  = ws + F_DALL;
  int*   sel   = (int*)(ws + F_SEL);

  // K0: Ds_all (960 rows)
  k_dall<<<4, 256, 0, stream>>>(pdw, Ds, Dall);

  // K1: conv 96->384 + bias + SiLU (WMMA f32). waves = 12 * 4608, 8 waves/block
  k_conv3x3<CIN, CH2, true><<<6912, 256, 0, stream>>>(x, w1, b1, xh, zbuf);

  // K2: cell means
  k_cells<<<B_*NCELL, 256, 0, stream>>>(xh, cells);

  // K3: top-10 threshold + prefix tables
  k_select<<<B_, 256, 0, stream>>>(cells, sel);

  // K4: gather selected pixels
  k_gather<<<(B_*HW)/256, 256, 0, stream>>>(xh, sel, xs);

  // K5a: x_dbl projection
  k_xdbl<<<(B_*K4*CPROJ*LSEL)/256, 256, 0, stream>>>(xs, xpw, xdbl);

  // K5b: dt projection
  k_dts<<<(B_*K4*DIN*LSEL)/256, 256, 0, stream>>>(xdbl, dtw, dts);

  // K6: selective scan (32 blocks of 192 threads)
  k_scan<<<B_*K4, DIN, 0, stream>>>(xs, xdbl, dts, dtb, alog, Y);

  // K7: combine directions
  k_comb<<<(B_*LSEL*DIN)/256, 256, 0, stream>>>(Y, ysc);

  // K8: assemble + LN + gate
  k_ln<<<B_*HW, DIN, 0, stream>>>(xh, zbuf, ysc, Dall, sel, lnw, lnb, ybuf);

  // K9: conv 192->96 (WMMA f32). waves = 3 * 4608
  k_conv3x3<DIN, CIN, false><<<1728, 256, 0, stream>>>(ybuf, w2, nullptr, out, nullptr);
}